// GraphPropagation_26207890440714
// MI455X (gfx1250) — compile-verified
//
#include <hip/hip_runtime.h>
#include <math.h>

#define KPARTS 6
#define BROWS  256
#define DDIM   256
#define NCOLS  50000
#define NT     32          // N-columns per block in the GEMM
#define LDS_STRIDE 260     // 256 + 4 floats pad -> conflict-free ds_load_b64
#define TOPK   5
#define TEMP   3.0f

typedef float v2f __attribute__((ext_vector_type(2)));
typedef float v4f __attribute__((ext_vector_type(4)));
typedef float v8f __attribute__((ext_vector_type(8)));
typedef unsigned int v4u __attribute__((ext_vector_type(4)));
typedef int v4i __attribute__((ext_vector_type(4)));
typedef int v8i __attribute__((ext_vector_type(8)));

// ---------------------------------------------------------------------------
// Kernel 1: inverse L2 norms. One wave (32 lanes) per row of 256 floats.
// ---------------------------------------------------------------------------
__global__ __launch_bounds__(256) void norm_kernel(
    const float* __restrict__ feat, const float* __restrict__ mem,
    float* __restrict__ invF, float* __restrict__ invM)
{
  const int wave = (blockIdx.x * 256 + threadIdx.x) >> 5;
  const int lane = threadIdx.x & 31;
  const int nrows = KPARTS * BROWS + KPARTS * NCOLS;
  if (wave >= nrows) return;

  const float* src;
  float* dst;
  if (wave < KPARTS * BROWS) {
    src = feat + (size_t)wave * DDIM;
    dst = invF + wave;
  } else {
    const int r = wave - KPARTS * BROWS;
    src = mem + (size_t)r * DDIM;
    dst = invM + r;
  }
  // 256 floats / 32 lanes = 8 floats per lane, two b128 loads
  float4 x = *(const float4*)(src + lane * 8);
  float4 y = *(const float4*)(src + lane * 8 + 4);
  float ss = x.x*x.x + x.y*x.y + x.z*x.z + x.w*x.w
           + y.x*y.x + y.y*y.y + y.z*y.z + y.w*y.w;
  #pragma unroll
  for (int off = 16; off > 0; off >>= 1)
    ss += __shfl_xor(ss, off, 32);
  if (lane == 0)
    *dst = 1.0f / fmaxf(sqrtf(ss), 1e-12f);
}

// ---------------------------------------------------------------------------
// Kernel 2: sim[k,b,n] = (feat[k,b,:] . mem[k,n,:]) * invF[k,b] * invM[k,n]
// Block: 256 threads (8 waves). Tile: M=256 (all of B) x N=32.
// Wave w: N-subtile = w&1 (16 cols), M-subtiles = (w>>1) + 4*t, t=0..3.
// f32 WMMA 16x16x4; K (=D) loop of 64 steps.
// Mem tile staged into LDS by the Tensor Data Mover (TDM) with hardware
// row padding (256-DWORD interval, 4-DWORD pad -> stride 260 floats).
// ---------------------------------------------------------------------------
__global__ __launch_bounds__(256) void sim_gemm_kernel(
    const float* __restrict__ feat, const float* __restrict__ mem,
    const float* __restrict__ invF, const float* __restrict__ invM,
    float* __restrict__ simOut)
{
  __shared__ float smem[NT * LDS_STRIDE];

  const int k   = blockIdx.y;
  const int n0  = blockIdx.x * NT;
  const int tid = threadIdx.x;

  const float* memk = mem + (size_t)k * NCOLS * DDIM;

#if defined(__gfx1250__) && __has_builtin(__builtin_amdgcn_tensor_load_to_lds) && __has_builtin(__builtin_amdgcn_s_wait_tensorcnt)
  // ---- TDM: DMA the [rows x 256] f32 tile into LDS with hardware padding ---
  if (tid < 32) {                              // one wave issues (TDM ignores EXEC)
    const unsigned lds_off = (unsigned)(size_t)&smem[0];     // low 32b = LDS byte offset
    const unsigned long long ga =
        (unsigned long long)(size_t)(memk + (size_t)n0 * DDIM);
    int rows = NCOLS - n0; if (rows > NT) rows = NT;          // uniform clamp

    v4u g0;
    g0[0] = 1u;                                   // count=1 (valid), load, user mode
    g0[1] = lds_off;                              // lds_addr
    g0[2] = (unsigned)(ga & 0xFFFFFFFFu);         // global_addr[31:0]
    g0[3] = (unsigned)((ga >> 32) & 0x01FFFFFFu)  // global_addr[56:32]
          | (2u << 30);                           // type = 2 ("image")

    v8i g1;
    g1[0] = (2 << 16)     // data_size = 4 bytes
          | (1 << 20)     // pad_enable
          | (7 << 22)     // pad_interval code 7 = every 256 DWORDs
          | (3 << 25);    // pad_amount  code 3 = 4 DWORDs (stride 260 floats)
    g1[1] = (DDIM & 0xFFFF) << 16;                // tensor_dim0[15:0]
    g1[2] = (int)(((unsigned)(NCOLS & 0xFFFF)) << 16); // tensor_dim0[31:16]=0 | tensor_dim1[15:0]
    g1[3] = (DDIM & 0xFFFF) << 16;                // tensor_dim1[31:16]=0 | tile_dim0=256
    g1[4] = rows & 0xFFFF;                        // tile_dim1 | tile_dim2=0
    g1[5] = DDIM;                                 // tensor_dim0_stride[31:0]
    g1[6] = 0;                                    // stride hi | tensor_dim1_stride lo (unused, 2D)
    g1[7] = 0;

    v4i g2 = {0, 0, 0, 0};
    v4i g3 = {0, 0, 0, 0};
    v8i g4 = {0, 0, 0, 0, 0, 0, 0, 0};            // extra descriptor words (unused, 2D)
    __builtin_amdgcn_tensor_load_to_lds(g0, g1, g2, g3, g4, 0);
    __builtin_amdgcn_s_wait_tensorcnt(0);
  }
  __syncthreads();
#else
  // ---- fallback: register-staged coalesced b128 loads (zero-fill OOB) -----
  #pragma unroll
  for (int j = 0; j < (NT * DDIM) / (256 * 4); ++j) {   // 8 iterations
    const int idx4 = j * 256 + tid;       // float4 index within tile
    const int rrow = idx4 >> 6;           // / (DDIM/4)
    const int c4   = idx4 & 63;
    float4 val = make_float4(0.f, 0.f, 0.f, 0.f);
    const int gr = n0 + rrow;
    if (gr < NCOLS)
      val = *(const float4*)(memk + (size_t)gr * DDIM + c4 * 4);
    *(float4*)(&smem[rrow * LDS_STRIDE + c4 * 4]) = val;
  }
  __syncthreads();
#endif

  const int lane  = tid & 31;
  const int w     = tid >> 5;
  const int lm    = lane & 15;
  const int lh    = lane >> 4;     // 0 or 1: selects K pair
  const int ntile = w & 1;
  const int mgrp  = w >> 1;        // 0..3

  const float* featk = feat + (size_t)k * BROWS * DDIM;
  const float* arow  = featk + (size_t)(mgrp * 16 + lm) * DDIM;   // tile t at +64*DDIM*t
  const float* bptr  = &smem[(ntile * 16 + lm) * LDS_STRIDE];

  v8f acc0 = {}, acc1 = {}, acc2 = {}, acc3 = {};

  #pragma unroll 4
  for (int k0 = 0; k0 < DDIM; k0 += 4) {
    const int kk = k0 + 2 * lh;
    v2f b  = *(const v2f*)(bptr + kk);                    // LDS b64, no bank conflict
    v2f a0 = *(const v2f*)(arow + kk);                    // L2-resident feat
    v2f a1 = *(const v2f*)(arow +  64 * DDIM + kk);
    v2f a2 = *(const v2f*)(arow + 128 * DDIM + kk);
    v2f a3 = *(const v2f*)(arow + 192 * DDIM + kk);
    acc0 = __builtin_amdgcn_wmma_f32_16x16x4_f32(false, a0, false, b, (short)0, acc0, false, false);
    acc1 = __builtin_amdgcn_wmma_f32_16x16x4_f32(false, a1, false, b, (short)0, acc1, false, false);
    acc2 = __builtin_amdgcn_wmma_f32_16x16x4_f32(false, a2, false, b, (short)0, acc2, false, false);
    acc3 = __builtin_amdgcn_wmma_f32_16x16x4_f32(false, a3, false, b, (short)0, acc3, false, false);
  }

  // ---- epilogue: scale by inverse norms, store (skip OOB columns) ----------
  const int ncol = n0 + ntile * 16 + lm;
  if (ncol < NCOLS) {
    const float  invMn = invM[(size_t)k * NCOLS + ncol];
    float*       outk  = simOut + (size_t)k * BROWS * NCOLS;
    const float* invFk = invF + k * BROWS;

#define STORE_TILE(ACC, T)                                                \
    {                                                                     \
      const int mbase = mgrp * 16 + (T) * 64 + 8 * lh;                    \
      _Pragma("unroll")                                                   \
      for (int r = 0; r < 8; ++r) {                                       \
        const int m = mbase + r;                                          \
        outk[(size_t)m * NCOLS + ncol] = ACC[r] * invFk[m] * invMn;       \
      }                                                                   \
    }
    STORE_TILE(acc0, 0)
    STORE_TILE(acc1, 1)
    STORE_TILE(acc2, 2)
    STORE_TILE(acc3, 3)
#undef STORE_TILE
  }
}

// ---------------------------------------------------------------------------
// Kernel 3: zero-fill soft_labels. Non-temporal stores: 307 MB of streaming
// zeros must not evict the L2-resident feat matrix / fresh sim lines.
// (Exact: the non-top-5 softmax terms underflow to 0 in fp32.)
// ---------------------------------------------------------------------------
__global__ __launch_bounds__(256) void zero_kernel(float* __restrict__ out, long n4)
{
  const long stride = (long)gridDim.x * blockDim.x;
  const v4f z = {0.f, 0.f, 0.f, 0.f};
  v4f* out4 = (v4f*)out;
  for (long i = (long)blockIdx.x * blockDim.x + threadIdx.x; i < n4; i += stride)
    __builtin_nontemporal_store(z, &out4[i]);
}

// ---------------------------------------------------------------------------
// Kernel 4: top-5 per row. One block per (k,b) row; register top-5 per
// thread, then LDS tree merge 256 -> 32 -> 1.
// ---------------------------------------------------------------------------
__device__ __forceinline__ void insert5(float (&v)[TOPK], int (&id)[TOPK],
                                        float x, int xi)
{
  if (x > v[TOPK - 1]) {
    v[TOPK - 1] = x; id[TOPK - 1] = xi;
    #pragma unroll
    for (int j = TOPK - 1; j > 0; --j) {
      if (v[j] > v[j - 1]) {
        float tv = v[j]; v[j] = v[j - 1]; v[j - 1] = tv;
        int   ti = id[j]; id[j] = id[j - 1]; id[j - 1] = ti;
      }
    }
  }
}

__global__ __launch_bounds__(256) void topk_kernel(
    const float* __restrict__ sim, float* __restrict__ topv, int* __restrict__ topi)
{
  __shared__ float sv[256 * TOPK];
  __shared__ int   si[256 * TOPK];

  const int row  = blockIdx.x;
  const int tid  = threadIdx.x;
  const float* p = sim + (size_t)row * NCOLS;

  float v[TOPK]; int id[TOPK];
  #pragma unroll
  for (int j = 0; j < TOPK; ++j) { v[j] = -3.4e38f; id[j] = 0; }

  for (int i = tid; i < NCOLS; i += 256)
    insert5(v, id, p[i], i);

  #pragma unroll
  for (int j = 0; j < TOPK; ++j) { sv[tid * TOPK + j] = v[j]; si[tid * TOPK + j] = id[j]; }
  __syncthreads();

  if (tid < 32) {
    float mv[TOPK]; int mi[TOPK];
    #pragma unroll
    for (int j = 0; j < TOPK; ++j) { mv[j] = -3.4e38f; mi[j] = 0; }
    for (int t = tid * 8; t < tid * 8 + 8; ++t)
      #pragma unroll
      for (int j = 0; j < TOPK; ++j)
        insert5(mv, mi, sv[t * TOPK + j], si[t * TOPK + j]);
    __syncthreads();
    #pragma unroll
    for (int j = 0; j < TOPK; ++j) { sv[tid * TOPK + j] = mv[j]; si[tid * TOPK + j] = mi[j]; }
  } else {
    __syncthreads();
  }
  __syncthreads();

  if (tid == 0) {
    float fv[TOPK]; int fi[TOPK];
    #pragma unroll
    for (int j = 0; j < TOPK; ++j) { fv[j] = -3.4e38f; fi[j] = 0; }
    for (int t = 0; t < 32; ++t)
      #pragma unroll
      for (int j = 0; j < TOPK; ++j)
        insert5(fv, fi, sv[t * TOPK + j], si[t * TOPK + j]);
    #pragma unroll
    for (int j = 0; j < TOPK; ++j) {
      topv[row * TOPK + j] = fv[j];
      topi[row * TOPK + j] = fi[j];
    }
  }
}

// ---------------------------------------------------------------------------
// Kernel 5: softmax over the 5 survivors, scatter into zeroed soft_labels
// ---------------------------------------------------------------------------
__global__ __launch_bounds__(256) void scatter_kernel(
    const float* __restrict__ topv, const int* __restrict__ topi,
    float* __restrict__ soft)
{
  const int row = blockIdx.x * blockDim.x + threadIdx.x;
  if (row >= KPARTS * BROWS) return;

  float v[TOPK]; int id[TOPK];
  #pragma unroll
  for (int j = 0; j < TOPK; ++j) { v[j] = topv[row * TOPK + j]; id[j] = topi[row * TOPK + j]; }

  float m = v[0];
  #pragma unroll
  for (int j = 1; j < TOPK; ++j) m = fmaxf(m, v[j]);

  float e[TOPK], s = 0.f;
  #pragma unroll
  for (int j = 0; j < TOPK; ++j) { e[j] = __expf((v[j] - m) / TEMP); s += e[j]; }
  const float inv = 1.0f / s;

  float* prow = soft + (size_t)row * NCOLS;
  #pragma unroll
  for (int j = 0; j < TOPK; ++j)
    prow[id[j]] = e[j] * inv;
}

// ---------------------------------------------------------------------------
extern "C" void kernel_launch(void* const* d_in, const int* in_sizes, int n_in,
                              void* d_out, int out_size, void* d_ws, size_t ws_size,
                              hipStream_t stream) {
  const float* feat = (const float*)d_in[0];   // [K,B,D]
  const float* mem  = (const float*)d_in[1];   // [K,N,D]

  float* soft   = (float*)d_out;                                    // [K,B,N]
  float* simOut = (float*)d_out + (size_t)KPARTS * BROWS * NCOLS;   // [K,B,N]

  // workspace layout (floats): invF[1536] | invM[300000] | topv[7680] | topi[7680]
  float* ws   = (float*)d_ws;
  float* invF = ws;
  float* invM = ws + KPARTS * BROWS;
  float* topv = invM + (size_t)KPARTS * NCOLS;
  int*   topi = (int*)(topv + KPARTS * BROWS * TOPK);

  // 1) inverse norms: 301536 rows, 8 waves per block
  const int nrows = KPARTS * BROWS + KPARTS * NCOLS;
  const int normBlocks = (nrows + 7) / 8;
  norm_kernel<<<normBlocks, 256, 0, stream>>>(feat, mem, invF, invM);

  // 2) WMMA GEMM -> sim
  dim3 g2((NCOLS + NT - 1) / NT, KPARTS);
  sim_gemm_kernel<<<g2, 256, 0, stream>>>(feat, mem, invF, invM, simOut);

  // 3) zero-fill soft_labels
  const long n4 = (long)KPARTS * BROWS * NCOLS / 4;
  zero_kernel<<<2048, 256, 0, stream>>>(soft, n4);

  // 4) top-5 per row
  topk_kernel<<<KPARTS * BROWS, 256, 0, stream>>>(simOut, topv, topi);

  // 5) softmax over survivors, scatter
  scatter_kernel<<<(KPARTS * BROWS + 255) / 256, 256, 0, stream>>>(topv, topi, soft);
}